// S4DRKernel_70068096467955
// MI455X (gfx1250) — compile-verified
//
#include <hip/hip_runtime.h>
#include <hip/hip_bf16.h>
#include <math.h>

// S4D Vandermonde kernel for MI455X (gfx1250), wave32 + WMMA f32 path.
//
// K[h, l] = 2 * Re( sum_n Cmod[h,n] * w[h,n]^l ),  w = exp(dt[h] * A[n])
// Tile l = 256*t + 16*i + j  ->  per (h, t):
//   K_tile(16x16) = ReA(16x64) x ReB(64x16)  +  (-ImA)(16x64) x ImB(64x16)
// as 16 chained v_wmma_f32_16x16x4_f32 pairs (K-chunks of 4).
//
// Transcendentals use hardware v_exp/v_sin/v_cos with an explicit
// revolutions-based range reduction (no libm Payne-Hanek in the hot loop):
//   w^p = exp(xr*p) * cis(2pi * frac(p * xi/2pi))
// B (= w^j, j<16) is shared by all 8 waves of a block -> built once in LDS.

typedef __attribute__((ext_vector_type(2))) float v2f;
typedef __attribute__((ext_vector_type(8))) float v8f;

#define NSTATE 64          // d_state (fixed by the reference)
#define TILE_L 256         // 16 rows x 16 cols of l per wave-tile
#define WAVES_PER_BLOCK 8  // 256 threads / wave32
#define TWO_PI 6.2831853071795864769f

__global__ __launch_bounds__(256) void s4d_wmma_f32_kernel(
    const float* __restrict__ log_dt,  // (H,)
    const float* __restrict__ C,       // (H, 64, 2) interleaved re/im
    const float* __restrict__ A,       // (64, 2) interleaved re/im
    float* __restrict__ K,             // (H, L)
    int L)
{
    const int h   = blockIdx.x;
    const int tid = threadIdx.x;

    // Per-h shared state
    __shared__ float s_xr[NSTATE];     // Re(dtA)
    __shared__ float s_r2[NSTATE];     // Im(dtA) / 2pi
    __shared__ float s_cre[NSTATE];    // Re(Cmod)
    __shared__ float s_cim[NSTATE];    // Im(Cmod)
    __shared__ float s_bre[NSTATE * 16];  // Re(w^j), [n*16+j]
    __shared__ float s_bim[NSTATE * 16];  // Im(w^j)

    // ---- stage 1: per-n parameters --------------------------------------
    if (tid < NSTATE) {
        const int n     = tid;
        const float dt  = __expf(log_dt[h]);
        const float are = A[2 * n + 0];
        const float aim = A[2 * n + 1];
        const float xre = dt * are;          // Re(dtA)
        const float xim = dt * aim;          // Im(dtA), |xim| small (< ~1.5)
        // w - 1 = exp(dtA) - 1   (same naive form as the reference)
        float sn, cs;
        __sincosf(xim, &sn, &cs);
        const float er    = __expf(xre);
        const float numre = er * cs - 1.0f;
        const float numim = er * sn;
        // (w-1)/A = (w-1)*conj(A)/|A|^2
        const float inv = 1.0f / (are * are + aim * aim);
        const float qre = (numre * are + numim * aim) * inv;
        const float qim = (numim * are - numre * aim) * inv;
        const float cre = C[(h * NSTATE + n) * 2 + 0];
        const float cim = C[(h * NSTATE + n) * 2 + 1];
        s_cre[n] = cre * qre - cim * qim;
        s_cim[n] = cre * qim + cim * qre;
        s_xr[n]  = xre;
        s_r2[n]  = xim * (1.0f / TWO_PI);
    }
    __syncthreads();

    // ---- stage 2: shared B matrix, w_n^j for j in [0,16) ----------------
    #pragma unroll
    for (int k = 0; k < (NSTATE * 16) / 256; ++k) {
        const int e = k * 256 + tid;       // e = n*16 + j
        const int n = e >> 4;
        const float j = (float)(e & 15);
        const float eb  = __expf(s_xr[n] * j);
        float rev = s_r2[n] * j;
        rev -= rintf(rev);
        float sb, cb;
        __sincosf(rev * TWO_PI, &sb, &cb);
        s_bre[e] = eb * cb;
        s_bim[e] = eb * sb;
    }
    __syncthreads();

    // ---- stage 3: WMMA tiles --------------------------------------------
    const int wave  = tid >> 5;        // which 256-wide l-tile
    const int lane  = tid & 31;
    const int row   = lane & 15;       // A-matrix M row / B-matrix column j
    const int khalf = (lane >> 4) * 2; // K sub-index: lanes 0-15 -> {0,1}, 16-31 -> {2,3}

    const int ntiles = L / TILE_L;
    for (int t = wave; t < ntiles; t += WAVES_PER_BLOCK) {
        // A row i contributes power p = 256*t + 16*i
        const float pA = (float)(TILE_L * t + 16 * row);

        v8f acc = {};
        #pragma unroll
        for (int c = 0; c < NSTATE / 4; ++c) {
            v2f a_re, a_imn, b_re, b_im;
            #pragma unroll
            for (int u = 0; u < 2; ++u) {
                const int n = 4 * c + khalf + u;
                // ---- A element: Cmod_n * w_n^pA  (Im negated for 2nd GEMM)
                const float ea = __expf(s_xr[n] * pA);
                float rev = s_r2[n] * pA;
                rev -= rintf(rev);
                float sa, ca;
                __sincosf(rev * TWO_PI, &sa, &ca);
                const float cre = s_cre[n];
                const float cim = s_cim[n];
                a_re[u]  = ea * (cre * ca - cim * sa);
                a_imn[u] = ea * (-cre * sa - cim * ca);
                // ---- B element from LDS
                b_re[u] = s_bre[n * 16 + row];
                b_im[u] = s_bim[n * 16 + row];
            }
            // D = ReA x ReB + D ; D = (-ImA) x ImB + D  (full f32 accumulate)
            acc = __builtin_amdgcn_wmma_f32_16x16x4_f32(
                false, a_re,  false, b_re, (short)0, acc, false, false);
            acc = __builtin_amdgcn_wmma_f32_16x16x4_f32(
                false, a_imn, false, b_im, (short)0, acc, false, false);
        }

        // C/D layout: VGPR v -> M = v (+8 for lanes 16-31), N = lane%16
        const int jcol  = lane & 15;
        const int rbase = (lane >> 4) * 8;
        const long base = (long)h * L + (long)TILE_L * t;
        #pragma unroll
        for (int v = 0; v < 8; ++v) {
            const int i = v + rbase;
            K[base + 16 * i + jcol] = 2.0f * acc[v];
        }
    }
}

extern "C" void kernel_launch(void* const* d_in, const int* in_sizes, int n_in,
                              void* d_out, int out_size, void* d_ws, size_t ws_size,
                              hipStream_t stream) {
    const float* log_dt = (const float*)d_in[0];  // (H,)
    const float* C      = (const float*)d_in[1];  // (H, N, 2)
    const float* A      = (const float*)d_in[2];  // (N, 2)
    (void)d_ws; (void)ws_size; (void)n_in;

    const int H = in_sizes[0];
    const int L = out_size / H;   // 2048

    s4d_wmma_f32_kernel<<<dim3(H), dim3(256), 0, stream>>>(
        log_dt, C, A, (float*)d_out, L);
}